// Model_22110491640546
// MI455X (gfx1250) — compile-verified
//
#include <hip/hip_runtime.h>
#include <math.h>

// ---------------------------------------------------------------------------
// DDSP-style synth for MI455X (gfx1250, wave32).
//  packed: (256, 464) f32   frequencies: (128,) f32   noise: (256, 32768) f32
//  out: (32, 1, 32768) f32
// Key algebraic facts used:
//  * radians is constant over frames -> osc_bank[b,h,t] = sin(rad[b,h]*(t+1))
//  * where(x>=pi,0) is a no-op (x clipped to [0,1])
//  * irfft(rfft(x,ortho)*H,ortho) == (1/N) IDFT(H * DFT(x)); done as a
//    16384-pt complex FFT fully resident in LDS
//  * noise row viewed as float2[16384] is exactly the even/odd complex packing
//    the half-size FFT needs -> TDM can DMA it straight into the FFT buffer
// CDNA5 usage:
//  * v_wmma_f32_16x16x4_f32 for the softmax-weighted matvec (f0)
//  * tensor_load_to_lds (TDM, TENSORcnt) ping-pong double buffering of the
//    128KB noise rows, overlapped with the 28 FFT stages (2x128KB LDS only
//    possible with CDNA5's 320KB WGP LDS)
// ---------------------------------------------------------------------------

#define NSAMP    32768
#define MFFT     16384
#define LOG2M    14
#define NEV      8
#define NHARM    8
#define NYQF     11025.0f
#define MINF0    (40.0f/11025.0f)
#define MAXF0    (3000.0f/11025.0f)
#define F0SPAN   (MAXF0 - MINF0)
#define PI_F     3.14159265358979323846f
#define TWO_PI_F 6.28318530717958647692f

typedef __attribute__((ext_vector_type(2))) float        v2f;
typedef __attribute__((ext_vector_type(8))) float        v8f;
typedef __attribute__((ext_vector_type(4))) unsigned int u32x4;
typedef __attribute__((ext_vector_type(4))) int          i32x4;
typedef __attribute__((ext_vector_type(8))) int          i32x8;

__device__ __forceinline__ float sigm(float x)    { return 1.0f/(1.0f + expf(-x)); }
__device__ __forceinline__ float clamp01(float x) { return fminf(fmaxf(x, 0.0f), 1.0f); }

#if __has_builtin(__builtin_amdgcn_tensor_load_to_lds) && \
    __has_builtin(__builtin_amdgcn_s_wait_tensorcnt)
#define USE_TDM 1
#else
#define USE_TDM 0
#endif

#if USE_TDM
// Generic LDS pointer = {shared aperture [63:32] | byte offset [31:0]}
__device__ __forceinline__ unsigned lds_off(const void* p)
{
    return (unsigned)(size_t)p;
}

// TDM DMA of one contiguous 32768-float row (128KB) global -> LDS.
// D# per CDNA5 ISA ch.8: 2D tensor, data_size=4B, tile 32768x1.
__device__ __forceinline__ void tdm_load_row(const float* __restrict__ gsrc,
                                             unsigned lds_byte_off)
{
    const unsigned long long ga = (unsigned long long)(const void*)gsrc;
    u32x4 g0;
    g0.x = 1u;                                            // count=1, user D#
    g0.y = lds_byte_off;                                  // lds_addr
    g0.z = (unsigned)(ga & 0xffffffffu);                  // global_addr[31:0]
    g0.w = (unsigned)((ga >> 32) & 0x01ffffffu)           // global_addr[56:32]
         | 0x80000000u;                                   // type = 2 ("image")
    i32x8 g1;
    g1[0] = (int)(2u << 16);                              // data_size=4B, mask=0
    g1[1] = (int)(32768u << 16);                          // tensor_dim0[15:0]
    g1[2] = (int)((32768u >> 16) | (1u << 16));           // tdim0 hi | tensor_dim1=1
    g1[3] = (int)(32768u << 16);                          // tile_dim0 = 32768
    g1[4] = 1;                                            // tile_dim1=1, tile_dim2=0
    g1[5] = 32768;                                        // tensor_dim0_stride lo
    g1[6] = 0;                                            // stride hi / dim1_stride
    g1[7] = 0;
    const i32x4 gz = {0, 0, 0, 0};
#if defined(__clang_major__) && (__clang_major__ >= 23)
    const i32x8 gz8 = {0, 0, 0, 0, 0, 0, 0, 0};
    __builtin_amdgcn_tensor_load_to_lds(g0, g1, gz, gz, gz8, 0);
#else
    __builtin_amdgcn_tensor_load_to_lds(g0, g1, gz, gz, 0);
#endif
}
#endif // USE_TDM

// ---------------------------------------------------------------------------
// Kernel 1: f0 = softmax(sigmoid(packed[:, :128])) @ frequencies via WMMA.
// One wave per block handles a 16-batch tile. freqs folded into A (documented
// 32-bit 16x4 A layout); B = all-ones so B's K/N lane mapping is irrelevant.
// Two V_WMMA_F32_16X16X4_F32 accumulator chains give numerator (sum e*f) and
// denominator (sum e) of the softmax-weighted dot product.
// ---------------------------------------------------------------------------
__global__ void k_f0(const float* __restrict__ packed,
                     const float* __restrict__ freqs,
                     float* __restrict__ f0dot)
{
    const int lane = threadIdx.x;          // 0..31 (wave32)
    const int base = blockIdx.x * 16;      // 16 batches per wave
    const int m    = lane & 15;            // row within tile
    const int koff = (lane >> 4) * 2;      // lanes 0-15: K=0,1  lanes 16-31: K=2,3

    v2f ones; ones.x = 1.0f; ones.y = 1.0f;
    v8f cn = {0.0f,0.0f,0.0f,0.0f,0.0f,0.0f,0.0f,0.0f};
    v8f cd = {0.0f,0.0f,0.0f,0.0f,0.0f,0.0f,0.0f,0.0f};

    const float* prow = packed + (size_t)(base + m) * 464;

    for (int k0 = 0; k0 < 128; k0 += 4) {
        const int k = k0 + koff;
        const float e0 = expf(sigm(prow[k    ]));
        const float e1 = expf(sigm(prow[k + 1]));
        v2f ad; ad.x = e0;            ad.y = e1;             // denominator A
        v2f an; an.x = e0*freqs[k];   an.y = e1*freqs[k+1];  // numerator A
        cn = __builtin_amdgcn_wmma_f32_16x16x4_f32(false, an, false, ones,
                                                   (short)0, cn, false, false);
        cd = __builtin_amdgcn_wmma_f32_16x16x4_f32(false, ad, false, ones,
                                                   (short)0, cd, false, false);
    }

    // C/D layout: lanes 0-15 vgpr r -> (M=r, N=lane); lanes 16-31 -> M=8+r.
    if (lane == 0) {
        #pragma unroll
        for (int r = 0; r < 8; ++r) f0dot[base + r] = cn[r] / cd[r];
    } else if (lane == 16) {
        #pragma unroll
        for (int r = 0; r < 8; ++r) f0dot[base + 8 + r] = cn[r] / cd[r];
    }
}

// ---------------------------------------------------------------------------
// Kernel 2: filtered-noise path, one group (8 events) per block, FFT in LDS,
// TDM ping-pong streaming of the noise rows.
// ---------------------------------------------------------------------------
__device__ __forceinline__ float filtH(const float* __restrict__ fenv, int k)
{
    // lin_interp(fenv[64] -> 16384) sampled at bin k (k in [0, 16383])
    float pos = ((float)k + 0.5f) * (64.0f/16384.0f) - 0.5f;
    pos = fminf(fmaxf(pos, 0.0f), 63.0f);
    const int   lo = (int)pos;
    const int   hi = min(lo + 1, 63);
    const float w  = pos - (float)lo;
    return fenv[lo]*(1.0f - w) + fenv[hi]*w;
}

__device__ void fft_inplace(float2* __restrict__ z, int tid, int nthr, float sgn)
{
    // radix-2 DIT, input bit-reversed, output natural order
    for (int s = 1; s <= LOG2M; ++s) {
        const int half = 1 << (s - 1);
        for (int tx = tid; tx < (MFFT >> 1); tx += nthr) {
            const int   j   = tx & (half - 1);
            const int   i0  = ((tx >> (s - 1)) << s) + j;
            const int   i1  = i0 + half;
            const float ang = sgn * (TWO_PI_F * (float)j / (float)(half << 1));
            float sw, cw; __sincosf(ang, &sw, &cw);
            const float2 a = z[i0];
            const float2 b = z[i1];
            const float tr = b.x*cw - b.y*sw;
            const float ti = b.x*sw + b.y*cw;
            z[i0] = make_float2(a.x + tr, a.y + ti);
            z[i1] = make_float2(a.x - tr, a.y - ti);
        }
        __syncthreads();
    }
}

__global__ void k_fft(const float* __restrict__ packed,
                      const float* __restrict__ noise,
                      float* __restrict__ out)
{
    extern __shared__ float2 zb[];         // 2 x 16384 complex = 256 KB LDS
    __shared__ float s_fenv[64];
    __shared__ float s_amp[128];

    const int g    = blockIdx.x;           // 0..31
    const int tid  = threadIdx.x;          // 0..1023
    const int nthr = blockDim.x;           // 1024

    float acc[32];
    #pragma unroll
    for (int q = 0; q < 32; ++q) acc[q] = 0.0f;

#if USE_TDM
    // preload event 0 into buffer 0 (wave 0 only issues the DMA)
    if (tid < 32)
        tdm_load_row(noise + (size_t)(g*NEV) * NSAMP, lds_off(&zb[0]));
#endif

    for (int e = 0; e < NEV; ++e) {
        const int b = g*NEV + e;
        const float* pk = packed + (size_t)b * 464;
        float2* z = zb + (e & 1) * MFFT;

        if (tid < 64)        s_fenv[tid]     = sigm(pk[400 + tid]);
        else if (tid < 192)  s_amp[tid - 64] = sigm(pk[256 + (tid - 64)])*2.0f - 1.0f;

#if USE_TDM
        if (tid < 32)
            __builtin_amdgcn_s_wait_tensorcnt(0);   // wave0: current row landed
        __syncthreads();                            // publish LDS to all waves;
                                                    // also: everyone finished
                                                    // with the other buffer
        if (tid < 32 && (e + 1) < NEV)              // kick DMA for next event,
            tdm_load_row(noise + (size_t)(b + 1) * NSAMP,   // overlaps the FFT
                         lds_off(&zb[((e + 1) & 1) * MFFT]));

        // noise_u = 2*x-1 in place (float2 pair == even/odd complex packing)
        for (int i = tid; i < MFFT; i += nthr) {
            const float2 v = z[i];
            z[i] = make_float2(v.x*2.0f - 1.0f, v.y*2.0f - 1.0f);
        }
        __syncthreads();
        // bit-reversal permutation (disjoint swaps) for the DIT forward pass
        for (int i = tid; i < MFFT; i += nthr) {
            const int j = (int)(__brev((unsigned)i) >> (32 - LOG2M));
            if (j > i) { const float2 t0 = z[i]; z[i] = z[j]; z[j] = t0; }
        }
        __syncthreads();
#else
        __syncthreads();
        // fallback: direct bit-reversed load from global
        const float* nb = noise + (size_t)b * NSAMP;
        for (int i = tid; i < MFFT; i += nthr) {
            const int j = (int)(__brev((unsigned)i) >> (32 - LOG2M));
            z[j] = make_float2(nb[2*i]*2.0f - 1.0f, nb[2*i + 1]*2.0f - 1.0f);
        }
        __syncthreads();
#endif

        fft_inplace(z, tid, nthr, -1.0f);  // forward

        // unpack real spectrum, apply H, repack — pair (k, M-k) per thread
        for (int k = tid; k <= MFFT/2; k += nthr) {
            if (k == 0) {
                const float2 Z0 = z[0];
                // X[0]=Re+Im, X[M]=Re-Im; H[M]=0 (pad) -> Y[M]=0
                const float Y0 = filtH(s_fenv, 0) * (Z0.x + Z0.y);
                z[0] = make_float2(0.5f*Y0, 0.5f*Y0);
            } else if (k == MFFT/2) {
                const float  Hk = filtH(s_fenv, k);  // Z'[M/2] = H * Z[M/2]
                const float2 Zk = z[k];
                z[k] = make_float2(Hk*Zk.x, Hk*Zk.y);
            } else {
                const int    kc = MFFT - k;
                const float2 Zk = z[k];
                const float2 Zc = z[kc];
                // E=(Zk+conj(Zc))/2 ; O=-i(Zk-conj(Zc))/2
                const float Er = 0.5f*(Zk.x + Zc.x), Ei = 0.5f*(Zk.y - Zc.y);
                const float dr = 0.5f*(Zk.x - Zc.x), di = 0.5f*(Zk.y + Zc.y);
                const float Or = di, Oi = -dr;
                float ts, tc;                         // t = cis(-pi*k/M)
                __sincosf(-PI_F * (float)k / (float)MFFT, &ts, &tc);
                const float tOr = tc*Or - ts*Oi;
                const float tOi = tc*Oi + ts*Or;
                // X[k]=E+tO ; X[M-k]=conj(E-tO)
                const float Hk = filtH(s_fenv, k);
                const float Hc = filtH(s_fenv, kc);
                const float Ypr = Hk*(Er + tOr),   Ypi = Hk*(Ei + tOi);
                const float Ycr = Hc*(Er - tOr),   Yci = Hc*(tOi - Ei);
                // Ey=(Yp+conj(Yc))/2 ; Oy=conj(t)*(Yp-conj(Yc))/2
                const float Eyr = 0.5f*(Ypr + Ycr), Eyi = 0.5f*(Ypi - Yci);
                const float qr  = 0.5f*(Ypr - Ycr), qi  = 0.5f*(Ypi + Yci);
                const float Oyr = tc*qr + ts*qi;
                const float Oyi = tc*qi - ts*qr;
                // Z'[k]=Ey+i*Oy ; Z'[M-k]=conj(Ey)+i*conj(Oy)
                z[k]  = make_float2(Eyr - Oyi, Eyi + Oyr);
                z[kc] = make_float2(Eyr + Oyi, Oyr - Eyi);
            }
        }
        __syncthreads();

        // in-place bit-reversal permutation, then inverse FFT (+twiddles)
        for (int i = tid; i < MFFT; i += nthr) {
            const int j = (int)(__brev((unsigned)i) >> (32 - LOG2M));
            if (j > i) { const float2 t0 = z[i]; z[i] = z[j]; z[j] = t0; }
        }
        __syncthreads();

        fft_inplace(z, tid, nthr, +1.0f);

        // filt = (ifft/M) * clip(lin_interp(amp),0,1); accumulate over events
        const float invM = 1.0f / (float)MFFT;
        #pragma unroll 4
        for (int q = 0; q < 32; ++q) {
            const int    t  = q*1024 + tid;
            const float2 zz = z[t >> 1];
            const float  y  = ((t & 1) ? zz.y : zz.x) * invM;
            float pos = ((float)t + 0.5f)*(128.0f/32768.0f) - 0.5f;
            pos = fminf(fmaxf(pos, 0.0f), 127.0f);
            const int   lo = (int)pos;
            const int   hi = min(lo + 1, 127);
            const float w  = pos - (float)lo;
            const float a  = clamp01(s_amp[lo]*(1.0f - w) + s_amp[hi]*w);
            acc[q] += y * a;
        }
    }

    // each event's filt is broadcast over 8 harmonics before the sum -> x8
    #pragma unroll 4
    for (int q = 0; q < 32; ++q)
        out[(size_t)g*NSAMP + q*1024 + tid] = 8.0f * acc[q];
}

// ---------------------------------------------------------------------------
// Kernel 3: harmonic oscillator bank, added on top of the noise term.
// Blocks: 32 groups x 8 sample-chunks. Envelope scan (128 frames) recomputed
// into LDS; osc_bank collapses to sin(rad*(t+1)) since f0 is frame-constant.
// ---------------------------------------------------------------------------
__global__ void k_mix(const float* __restrict__ packed,
                      const float* __restrict__ f0dot,
                      float* __restrict__ out)
{
    __shared__ float s_env[NEV][NHARM][129];  // +1 pad vs bank conflicts
    __shared__ float s_rad[NEV][NHARM];

    const int g     = blockIdx.x >> 3;
    const int chunk = blockIdx.x & 7;
    const int tid   = threadIdx.x;            // 0..255

    if (tid < NEV*NHARM) {
        const int e = tid >> 3, h = tid & 7;
        const int b = g*NEV + e;
        const float* pk = packed + (size_t)b * 464;
        const float hamp = sigm(pk[384 + h]);
        const float hdec = 0.9f + 0.1f*sigm(pk[392 + h]);
        float cur = 0.0f;
        for (int i = 0; i < 128; ++i) {
            const float a = sigm(pk[256 + i])*2.0f - 1.0f;
            cur = clamp01(cur + a*hamp);
            s_env[e][h][i] = cur;             // value before decay
            cur *= hdec;
        }
        const float f0hz = f0dot[b] * NYQF;
        const float ff   = MINF0 + f0hz * F0SPAN;
        s_rad[e][h] = ff * (float)(h + 1) * PI_F / NYQF;
    }
    __syncthreads();

    const int t0 = chunk * 4096;
    for (int t = t0 + tid; t < t0 + 4096; t += 256) {
        float pos = ((float)t + 0.5f)*(128.0f/32768.0f) - 0.5f;
        pos = fminf(fmaxf(pos, 0.0f), 127.0f);
        const int   lo  = (int)pos;
        const int   hi  = min(lo + 1, 127);
        const float w   = pos - (float)lo;
        const float tp1 = (float)(t + 1);
        float sum = 0.0f;
        #pragma unroll
        for (int e = 0; e < NEV; ++e) {
            #pragma unroll
            for (int h = 0; h < NHARM; ++h) {
                const float env = s_env[e][h][lo]*(1.0f - w) + s_env[e][h][hi]*w;
                sum += env * sinf(s_rad[e][h] * tp1);
            }
        }
        out[(size_t)g*NSAMP + t] += sum;
    }
}

// ---------------------------------------------------------------------------
extern "C" void kernel_launch(void* const* d_in, const int* in_sizes, int n_in,
                              void* d_out, int out_size, void* d_ws, size_t ws_size,
                              hipStream_t stream)
{
    (void)in_sizes; (void)n_in; (void)out_size; (void)ws_size;
    const float* packed = (const float*)d_in[0];   // (256, 464)
    const float* freqs  = (const float*)d_in[1];   // (128,)
    const float* noise  = (const float*)d_in[2];   // (256, 32768)
    float*       out    = (float*)d_out;           // (32, 1, 32768)
    float*       f0dot  = (float*)d_ws;            // 256 floats scratch

    k_f0 <<<16,  32,  0,                          stream>>>(packed, freqs, f0dot);
    k_fft<<<32,  1024, 2*sizeof(float2)*MFFT,     stream>>>(packed, noise, out);
    k_mix<<<256, 256, 0,                          stream>>>(packed, f0dot, out);
}